// Network_41326175322490
// MI455X (gfx1250) — compile-verified
//
#include <hip/hip_runtime.h>
#include <stdint.h>

// ROI crop-and-resize (bilinear, 14x14) + 2x2 maxpool -> [1024, 512, 7, 7] f32.
// Memory-movement bound: 103 MB output stream (~4.4 us at 23.3 TB/s HBM),
// 4.9 MB feature map is L2-resident. CDNA5 paths used:
//   * TENSOR_LOAD_TO_LDS (TDM, TENSORcnt) for the 38,304 B channel-plane tile
//   * GLOBAL_LOAD_ASYNC_TO_LDS_B128 (ASYNCcnt) for the ROI table
// Random bilinear taps are then served from LDS; output via non-temporal stores.

namespace {
constexpr int   kC        = 512;
constexpr int   kH        = 38;
constexpr int   kW        = 63;
constexpr int   kHW       = kH * kW;          // 2394
constexpr int   kN        = 1024;
constexpr int   kPool     = 7;
constexpr int   kCells    = kPool * kPool;    // 49
constexpr float kScale    = 1.0f / 16.0f;

constexpr int   kCPB      = 4;                // contiguous channels per block
constexpr int   kRPB      = 64;               // rois per block
constexpr int   kThreads  = 256;              // 8 waves (wave32)
constexpr int   kChChunks = kC / kCPB;        // 128
constexpr int   kRoiChunk = kN / kRPB;        // 16
constexpr int   kFeatElems = kCPB * kHW;      // 9576 floats = 38,304 B
constexpr int   kFeatB128  = (kFeatElems * 4) / 16;   // 2394
constexpr int   kRoiB128   = (kRPB * 5 * 4) / 16;     // 80
} // namespace

typedef int      v4i  __attribute__((vector_size(16)));
typedef unsigned u32x4 __attribute__((vector_size(16)));
typedef int      i32x4 __attribute__((vector_size(16)));
typedef int      i32x8 __attribute__((vector_size(32)));

#if defined(__HIP_DEVICE_COMPILE__)
#if __has_builtin(__builtin_amdgcn_global_load_async_to_lds_b128)
#define USE_ASYNC_LDS 1
#endif
#if __has_builtin(__builtin_amdgcn_tensor_load_to_lds) && defined(USE_ASYNC_LDS)
#define USE_TDM 1
#endif
#endif

__global__ __launch_bounds__(kThreads)
void roi_crop_maxpool_kernel(const float* __restrict__ bottom,
                             const float* __restrict__ rois,
                             float* __restrict__ out) {
  __shared__ __align__(16) float sfeat[kFeatElems];   // 38,304 B feature tile
  __shared__ __align__(16) float sroi[kRPB * 5];      // raw roi rows (1,280 B)

  const int tid  = threadIdx.x;
  const int cc   = (int)blockIdx.x % kChChunks;       // channel chunk
  const int rc   = (int)blockIdx.x / kChChunks;       // roi chunk
  const int roi0 = rc * kRPB;

  const float* gsrc = bottom + (size_t)cc * kFeatElems; // 16B-aligned (38304%16==0)
  const float* rsrc = rois + (size_t)roi0 * 5;          // 16B-aligned (1280%16==0)

#if defined(USE_ASYNC_LDS)
  // --- ROI table: per-lane async global->LDS b128 copies (ASYNCcnt). ---
  {
    auto grb = (__attribute__((address_space(1))) v4i*)(uintptr_t)rsrc;
    auto lrb = (__attribute__((address_space(3))) v4i*)(uint32_t)(uintptr_t)(&sroi[0]);
    for (int i = tid; i < kRoiB128; i += kThreads) {
      __builtin_amdgcn_global_load_async_to_lds_b128(grb + i, lrb + i, 0, 0);
    }
  }

#if defined(USE_TDM)
  // --- Feature tile: one TDM descriptor, issued by wave 0 (TENSORcnt). ---
  if (tid < 32) {
    const uint32_t lds_addr = (uint32_t)(uintptr_t)(&sfeat[0]);
    const uint64_t ga       = (uint64_t)(uintptr_t)gsrc;
    // D# group0: count=1 | lds_addr | global_addr[56:0] | type=2
    u32x4 g0 = { 1u, lds_addr, (unsigned)(ga & 0xFFFFFFFFu),
                 (unsigned)((ga >> 32) & 0x01FFFFFFu) | (2u << 30) };
    // D# group1: data_size=4B; tensor_dim0=tile_dim0=9576; dim1=1; stride0=9576.
    i32x8 g1 = { (int)0x00020000,            // wg_mask=0, data_size=2 (4B)
                 (int)(9576u << 16),         // [31:16] tensor_dim0 lo16
                 (int)(1u << 16),            // [31:16] tensor_dim1 lo16, dim0 hi16=0
                 (int)(9576u << 16),         // [31:16] tile_dim0, dim1 hi16=0
                 (int)1,                     // [15:0] tile_dim1=1, tile_dim2=0
                 (int)9576,                  // tensor_dim0_stride lo32
                 (int)(9576u << 16),         // [31:16] tensor_dim1_stride lo16
                 0 };                        // tensor_dim1_stride hi32
    i32x4 gz = { 0, 0, 0, 0 };               // groups 2/3 unused (<=2D tensor)
#if __clang_major__ >= 23
    i32x8 z8 = { 0, 0, 0, 0, 0, 0, 0, 0 };
    __builtin_amdgcn_tensor_load_to_lds(g0, g1, gz, gz, z8, 0);
#else
    __builtin_amdgcn_tensor_load_to_lds(g0, g1, gz, gz, 0);
#endif
  }
#else
  // --- Fallback: feature tile via per-lane async b128 copies. ---
  {
    auto gfb = (__attribute__((address_space(1))) v4i*)(uintptr_t)gsrc;
    auto lfb = (__attribute__((address_space(3))) v4i*)(uint32_t)(uintptr_t)(&sfeat[0]);
    for (int i = tid; i < kFeatB128; i += kThreads) {
      __builtin_amdgcn_global_load_async_to_lds_b128(gfb + i, lfb + i, 0, 0);
    }
  }
#endif

  // Wait for this wave's async copies; wave 0 also drains its tensor DMA.
#if __has_builtin(__builtin_amdgcn_s_wait_asynccnt)
  __builtin_amdgcn_s_wait_asynccnt(0);
#else
  asm volatile("s_wait_asynccnt 0" ::: "memory");
#endif
#if defined(USE_TDM)
  if (tid < 32) {
    __builtin_amdgcn_s_wait_tensorcnt(0);
  }
#endif

#else  // !USE_ASYNC_LDS: plain synchronous staging (host pass / old toolchain)
  {
    const float4* g4 = (const float4*)gsrc;
    float4*       s4 = (float4*)sfeat;
    for (int i = tid; i < kFeatB128; i += kThreads) s4[i] = g4[i];
    const float4* r4 = (const float4*)rsrc;
    float4*       q4 = (float4*)sroi;
    for (int i = tid; i < kRoiB128; i += kThreads) q4[i] = r4[i];
  }
#endif
  __syncthreads();

  // Each thread produces pooled outputs; grid coords recomputed in registers
  // (cheap VALU) so no barriers inside the ROI loop.
  constexpr int kPerRoi = kCPB * kCells;      // 196
  constexpr int kTotal  = kRPB * kPerRoi;     // 12544
  constexpr float kInv13 = 1.0f / 13.0f;      // t = grid_idx / (14-1)

  for (int idx = tid; idx < kTotal; idx += kThreads) {
    const int r_local = idx / kPerRoi;
    const int rem     = idx - r_local * kPerRoi;
    const int c_local = rem / kCells;
    const int cell    = rem - c_local * kCells;
    const int py      = cell / kPool;
    const int px      = cell - py * kPool;

    const float* rp  = &sroi[r_local * 5];
    const float x1s = rp[1] * kScale;
    const float y1s = rp[2] * kScale;
    const float x2s = rp[3] * kScale;
    const float y2s = rp[4] * kScale;
    const float dx  = (x2s - x1s) * kInv13;
    const float dy  = (y2s - y1s) * kInv13;

    const float* plane = &sfeat[c_local * kHW];

    float m = -3.402823466e+38f;

#pragma unroll
    for (int sy = 0; sy < 2; ++sy) {
      const float in_y = y1s + (float)(2 * py + sy) * dy;
      const bool  vy   = (in_y >= 0.0f) && (in_y <= (float)(kH - 1));
      const float y0f  = floorf(in_y);
      const float wy   = in_y - y0f;
      const int   yi   = (int)y0f;
      const int   y0   = min(max(yi, 0), kH - 1);
      const int   y1c  = min(max(yi + 1, 0), kH - 1);
      const float* row0 = plane + y0 * kW;
      const float* row1 = plane + y1c * kW;
#pragma unroll
      for (int sx = 0; sx < 2; ++sx) {
        const float in_x = x1s + (float)(2 * px + sx) * dx;
        const bool  vx   = (in_x >= 0.0f) && (in_x <= (float)(kW - 1));
        const float x0f  = floorf(in_x);
        const float wx   = in_x - x0f;
        const int   xi   = (int)x0f;
        const int   x0   = min(max(xi, 0), kW - 1);
        const int   x1c  = min(max(xi + 1, 0), kW - 1);

        const float g00 = row0[x0];
        const float g01 = row0[x1c];
        const float g10 = row1[x0];
        const float g11 = row1[x1c];

        const float top = g00 + (g01 - g00) * wx;
        const float bot = g10 + (g11 - g10) * wx;
        float       v   = top + (bot - top) * wy;
        v = (vx && vy) ? v : 0.0f;
        m = fmaxf(m, v);
      }
    }

    const int    roi = roi0 + r_local;
    const int    c   = cc * kCPB + c_local;
    const size_t o   = ((size_t)roi * kC + c) * kCells + cell;
    __builtin_nontemporal_store(m, &out[o]);   // write-once 103 MB stream
  }
}

extern "C" void kernel_launch(void* const* d_in, const int* in_sizes, int n_in,
                              void* d_out, int out_size, void* d_ws, size_t ws_size,
                              hipStream_t stream) {
  (void)in_sizes; (void)n_in; (void)out_size; (void)d_ws; (void)ws_size;
  const float* bottom = (const float*)d_in[0];   // [1,512,38,63] f32
  const float* rois   = (const float*)d_in[1];   // [1024,5]      f32
  float*       out    = (float*)d_out;           // [1024,512,7,7] f32

  dim3 grid(kChChunks * kRoiChunk);              // 2048 blocks
  dim3 block(kThreads);                          // 8 waves of 32
  hipLaunchKernelGGL(roi_crop_maxpool_kernel, grid, block, 0, stream,
                     bottom, rois, out);
}